// attentiona_21449066676905
// MI455X (gfx1250) — compile-verified
//
#include <hip/hip_runtime.h>
#include <hip/hip_bf16.h>
#include <math.h>

typedef __attribute__((ext_vector_type(2))) float v2f;
typedef __attribute__((ext_vector_type(8))) float v8f;

// Problem constants
#define NB    8
#define NCTX  4096
#define NDIM  1024
#define NHEAD 16
#define NHD   64
// diff denominator = B*HEAD*CTX*HD
#define DIFF_DENOM 33554432.0f

// ws layout, offsets in floats
#define XA_OFF  0u         // LN(x,lna) active rows  [128][1024]
#define XB_OFF  131072u    // LN(x,lnb) active rows  [128][1024]
#define QC_OFF  262144u    // qcur  [bh=128][16][64]
#define KH_OFF  393216u    // k head tiles
#define VH_OFF  524288u    // v head tiles
#define KN_OFF  655360u    // kn
#define VI_OFF  786432u    // vi
#define IT_OFF  917504u    // iter_out
#define PV_OFF  1048576u   // prev
#define AO_OFF  1179648u   // attn_out
#define CT_OFF  1310720u   // control: f[0..2]=diff accum; int slots at +8: done, upd_attn, upd_q

// ---------------------------------------------------------------- fill output
__global__ void k_fill(float* __restrict__ out, const float* __restrict__ ob,
                       float* __restrict__ ws) {
  const float4* ob4 = (const float4*)ob;
  float4* o4 = (float4*)out;
  size_t total4 = (size_t)NB * NCTX * NDIM / 4;   // 8388608
  for (size_t i = (size_t)blockIdx.x * blockDim.x + threadIdx.x; i < total4;
       i += (size_t)gridDim.x * blockDim.x) {
    o4[i] = ob4[i & 255u];   // broadcast ob over every row
  }
  if (blockIdx.x == 0 && threadIdx.x < 16) ws[CT_OFF + threadIdx.x] = 0.0f;
}

// ---------------------------------------------------------------- LayerNorm of 128 active rows
__global__ void k_ln(const float* __restrict__ x, const float* __restrict__ lna,
                     const float* __restrict__ lnb, float* __restrict__ ws) {
  int r = blockIdx.x;                 // 0..127 == b*16+s
  int b = r >> 4, s = r & 15;
  const float* xr = x + ((size_t)b * NCTX + s) * NDIM;
  int t = threadIdx.x;                // 256 threads, 4 elems each
  float4 xv = ((const float4*)xr)[t];
  __shared__ float rs[256], rq[256];
  rs[t] = xv.x + xv.y + xv.z + xv.w;
  rq[t] = xv.x * xv.x + xv.y * xv.y + xv.z * xv.z + xv.w * xv.w;
  __syncthreads();
  for (int off = 128; off > 0; off >>= 1) {
    if (t < off) { rs[t] += rs[t + off]; rq[t] += rq[t + off]; }
    __syncthreads();
  }
  float mean = rs[0] * (1.0f / NDIM);
  float var  = rq[0] * (1.0f / NDIM) - mean * mean;
  float rstd = rsqrtf(var + 1e-5f);
  float* xa  = ws + XA_OFF + (size_t)r * NDIM;
  float* xbn = ws + XB_OFF + (size_t)r * NDIM;
  float v[4] = {xv.x, xv.y, xv.z, xv.w};
  for (int i = 0; i < 4; ++i) {
    int j = t * 4 + i;
    float nv = (v[i] - mean) * rstd;
    xa[j]  = nv * lna[j];
    xbn[j] = nv * lnb[j];
  }
}

// ---------------------------------------------------------------- q/k/v GEMMs via f32 WMMA
// grid = 192: id = nch*24 + mat*8 + mt ; block = 256 (8 waves), 1 tile/wave
__global__ void k_qkv(const float* __restrict__ qw, const float* __restrict__ qb,
                      const float* __restrict__ kw, const float* __restrict__ vw,
                      const float* __restrict__ vb, float* __restrict__ ws) {
  int id = blockIdx.x;
  int mt  = id & 7;          // m-tile == batch b
  int rem = id >> 3;
  int mat = rem % 3;         // 0=q 1=k 2=v
  int nch = rem / 3;         // 0..7
  int wave = threadIdx.x >> 5, lane = threadIdx.x & 31;
  int hsel = lane >> 4;      // K-pair select
  int l15  = lane & 15;
  int ntile = nch * 8 + wave;

  const float* A = ws + (mat == 0 ? XA_OFF : XB_OFF) +
                   ((size_t)(mt * 16 + l15)) * NDIM + hsel * 2;
  const float* W = (mat == 0) ? qw : (mat == 1) ? kw : vw;
  const float* Brow = W + ((size_t)(ntile * 16 + l15)) * NDIM + hsel * 2;

  v8f acc = {};
  for (int k = 0; k < NDIM; k += 4) {
    v2f af = *(const v2f*)(A + k);
    v2f bf = *(const v2f*)(Brow + k);
    acc = __builtin_amdgcn_wmma_f32_16x16x4_f32(false, af, false, bf,
                                                (short)0, acc, false, false);
  }

  int gn = ntile * 16 + l15;
  int h = gn >> 6, d = gn & 63;
  const float* bias = (mat == 0) ? qb : (mat == 2) ? vb : nullptr;
  float bv = bias ? bias[gn] : 0.0f;
  float* dst = ws + (mat == 0 ? QC_OFF : (mat == 1 ? KH_OFF : VH_OFF));
  for (int r = 0; r < 8; ++r) {
    int s = r + hsel * 8;                       // row within tile == seq pos
    dst[(((size_t)mt * 16 + h) * 16 + s) * 64 + d] = acc[r] + bv;
  }
}

// ---------------------------------------------------------------- kn, vi (iteration-invariant)
__global__ void k_kv(const float* __restrict__ lkw, const float* __restrict__ lkb,
                     const float* __restrict__ lvw, const float* __restrict__ lvb,
                     const float* __restrict__ lnd, float* __restrict__ ws) {
  int bh = blockIdx.x, t = threadIdx.x;
  __shared__ float Ks[1024], Vs[1024], Kt[1024];
  __shared__ float mu[16], rstd[16];
  size_t base = (size_t)bh * 1024;
  ((float4*)Ks)[t] = ((const float4*)(ws + KH_OFF + base))[t];
  ((float4*)Vs)[t] = ((const float4*)(ws + VH_OFF + base))[t];
  __syncthreads();
  int s = t >> 4, d0 = (t & 15) * 4;
  float* vi = ws + VI_OFF + base;
  for (int i = 0; i < 4; ++i) {
    int d = d0 + i;
    float ak = lkb[d], av = lvb[d];
    const float* wk = lkw + d * 64;
    const float* wv = lvw + d * 64;
    for (int j = 0; j < 64; ++j) { ak += Ks[s * 64 + j] * wk[j]; av += Vs[s * 64 + j] * wv[j]; }
    Kt[s * 64 + d] = ak;
    vi[s * 64 + d] = av;
  }
  __syncthreads();
  if (t < 16) {
    float sm = 0.f, sq = 0.f;
    for (int j = 0; j < 64; ++j) { float v = Kt[t * 64 + j]; sm += v; sq += v * v; }
    float m = sm * (1.0f / 64.0f);
    mu[t] = m; rstd[t] = rsqrtf(sq * (1.0f / 64.0f) - m * m + 1e-5f);
  }
  __syncthreads();
  float* kn = ws + KN_OFF + base;
  for (int i = 0; i < 4; ++i) {
    int d = d0 + i;
    kn[s * 64 + d] = (Kt[s * 64 + d] - mu[s]) * rstd[s] * lnd[d];
  }
}

// ---------------------------------------------------------------- one attention iteration
__global__ void k_attn(const float* __restrict__ lqw, const float* __restrict__ lqb,
                       const float* __restrict__ lnc, float* __restrict__ ws, int it) {
  int bh = blockIdx.x, t = threadIdx.x;
  __shared__ float Q[1024], QN[1024], KN[1024], VIs[1024], L[256];
  __shared__ float mu[16], rstd[16], red[256];
  size_t base = (size_t)bh * 1024;
  ((float4*)Q)[t]   = ((const float4*)(ws + QC_OFF + base))[t];
  ((float4*)KN)[t]  = ((const float4*)(ws + KN_OFF + base))[t];
  ((float4*)VIs)[t] = ((const float4*)(ws + VI_OFF + base))[t];
  __syncthreads();
  int s = t >> 4, d0 = (t & 15) * 4;
  for (int i = 0; i < 4; ++i) {
    int d = d0 + i;
    float a = lqb[d];
    const float* wq = lqw + d * 64;
    for (int j = 0; j < 64; ++j) a += Q[s * 64 + j] * wq[j];
    QN[s * 64 + d] = a;
  }
  __syncthreads();
  if (t < 16) {
    float sm = 0.f, sq = 0.f;
    for (int j = 0; j < 64; ++j) { float v = QN[t * 64 + j]; sm += v; sq += v * v; }
    float m = sm * (1.0f / 64.0f);
    mu[t] = m; rstd[t] = rsqrtf(sq * (1.0f / 64.0f) - m * m + 1e-5f);
  }
  __syncthreads();
  for (int i = 0; i < 4; ++i) {
    int d = d0 + i;
    QN[s * 64 + d] = (QN[s * 64 + d] - mu[s]) * rstd[s] * lnc[d];
  }
  __syncthreads();
  { // logits: one entry per thread
    int sq_ = t >> 4, sk = t & 15;
    float a = 0.f;
    for (int j = 0; j < 64; ++j) a += QN[sq_ * 64 + j] * KN[sk * 64 + j];
    L[t] = a * 0.125f;     // 1/sqrt(64)
  }
  __syncthreads();
  if (t < 16) { // softmax row t
    float mx = L[t * 16];
    for (int j = 1; j < 16; ++j) mx = fmaxf(mx, L[t * 16 + j]);
    float se = 0.f;
    for (int j = 0; j < 16; ++j) { float e = __expf(L[t * 16 + j] - mx); L[t * 16 + j] = e; se += e; }
    float inv = 1.0f / se;
    for (int j = 0; j < 16; ++j) L[t * 16 + j] *= inv;
  }
  __syncthreads();
  float part = 0.f;
  float* iter = ws + IT_OFF + base;
  const float* prev = ws + PV_OFF + base;
  for (int i = 0; i < 4; ++i) {
    int d = d0 + i;
    float a = 0.f;
    for (int sk = 0; sk < 16; ++sk) a += L[s * 16 + sk] * VIs[sk * 64 + d];
    float pv = (it == 0) ? 0.f : prev[s * 64 + d];
    part += fabsf(a - pv);
    iter[s * 64 + d] = a;
  }
  red[t] = part;
  __syncthreads();
  for (int off = 128; off > 0; off >>= 1) {
    if (t < off) red[t] += red[t + off];
    __syncthreads();
  }
  if (t == 0) atomicAdd(ws + CT_OFF + it, red[0]);
}

// ---------------------------------------------------------------- loop control (reference where() semantics)
__global__ void k_ctrl(float* __restrict__ ws, int it) {
  if (threadIdx.x == 0 && blockIdx.x == 0) {
    float diff = ws[CT_OFF + it] * (1.0f / DIFF_DENOM);
    int* ci = (int*)(ws + CT_OFF + 8);   // [0]=done [1]=upd_attn [2]=upd_q
    int done = (it == 0) ? 0 : ci[0];
    int brk = (diff < 0.01f + 0.1f * diff) && (it > 0) && (!done);
    ci[1] = !done;
    ci[2] = !(done || brk);
    ci[0] = done || brk;
  }
}

// ---------------------------------------------------------------- apply masked updates
__global__ void k_upd(float* __restrict__ ws) {
  const int* ci = (const int*)(ws + CT_OFF + 8);
  int ua = ci[1], uq = ci[2];
  int idx = blockIdx.x * blockDim.x + threadIdx.x;   // 128*256 = 32768 float4s
  float4 v = ((const float4*)(ws + IT_OFF))[idx];
  if (ua) {
    ((float4*)(ws + AO_OFF))[idx] = v;
    ((float4*)(ws + PV_OFF))[idx] = v;
  }
  if (uq) {
    float4 q = ((float4*)(ws + QC_OFF))[idx];
    q.x += v.x; q.y += v.y; q.z += v.z; q.w += v.w;
    ((float4*)(ws + QC_OFF))[idx] = q;
  }
}

// ---------------------------------------------------------------- final projection of 128 active rows
// grid = 64: id = nch*8 + mt ; 8 waves, one 16x16 tile each
__global__ void k_out(const float* __restrict__ ow, const float* __restrict__ ob,
                      const float* __restrict__ ws, float* __restrict__ out) {
  int id = blockIdx.x;
  int mt = id & 7;           // batch b
  int nch = id >> 3;
  int wave = threadIdx.x >> 5, lane = threadIdx.x & 31;
  int hsel = lane >> 4, l15 = lane & 15;
  int ntile = nch * 8 + wave;

  // A row (b=mt, s=l15) gathered from head layout attn_out[b][h][s][d]
  const float* Abase = ws + AO_OFF + (size_t)mt * 16384 + (size_t)l15 * 64;
  const float* Brow  = ow + ((size_t)(ntile * 16 + l15)) * NDIM + hsel * 2;

  v8f acc = {};
  for (int k = 0; k < NDIM; k += 4) {
    int kk = k + hsel * 2;
    int aoff = ((kk >> 6) << 10) | (kk & 63);   // h*1024 + d
    v2f af = *(const v2f*)(Abase + aoff);
    v2f bf = *(const v2f*)(Brow + k);
    acc = __builtin_amdgcn_wmma_f32_16x16x4_f32(false, af, false, bf,
                                                (short)0, acc, false, false);
  }
  int gn = ntile * 16 + l15;
  float bv = ob[gn];
  for (int r = 0; r < 8; ++r) {
    int s = r + hsel * 8;
    out[((size_t)mt * NCTX + s) * NDIM + gn] = acc[r] + bv;
  }
}

// ---------------------------------------------------------------- host launch
extern "C" void kernel_launch(void* const* d_in, const int* in_sizes, int n_in,
                              void* d_out, int out_size, void* d_ws, size_t ws_size,
                              hipStream_t stream) {
  (void)in_sizes; (void)n_in; (void)out_size; (void)ws_size;
  const float* x   = (const float*)d_in[0];
  const float* qw  = (const float*)d_in[1];
  const float* qb  = (const float*)d_in[2];
  const float* kw  = (const float*)d_in[3];
  const float* vw  = (const float*)d_in[4];
  const float* vb  = (const float*)d_in[5];
  const float* ow  = (const float*)d_in[6];
  const float* ob  = (const float*)d_in[7];
  const float* lna = (const float*)d_in[8];
  const float* lnb = (const float*)d_in[9];
  const float* lnc = (const float*)d_in[10];
  const float* lnd = (const float*)d_in[11];
  const float* lqw = (const float*)d_in[12];
  const float* lqb = (const float*)d_in[13];
  const float* lkw = (const float*)d_in[14];
  const float* lkb = (const float*)d_in[15];
  const float* lvw = (const float*)d_in[16];
  const float* lvb = (const float*)d_in[17];
  float* out = (float*)d_out;
  float* ws  = (float*)d_ws;

  k_fill<<<8192, 256, 0, stream>>>(out, ob, ws);
  k_ln<<<128, 256, 0, stream>>>(x, lna, lnb, ws);
  k_qkv<<<192, 256, 0, stream>>>(qw, qb, kw, vw, vb, ws);
  k_kv<<<128, 256, 0, stream>>>(lkw, lkb, lvw, lvb, lnd, ws);
  for (int it = 0; it < 3; ++it) {
    k_attn<<<128, 256, 0, stream>>>(lqw, lqb, lnc, ws, it);
    k_ctrl<<<1, 32, 0, stream>>>(ws, it);
    k_upd<<<128, 256, 0, stream>>>(ws);
  }
  k_out<<<64, 256, 0, stream>>>(ow, ob, ws, out);
}